// COPNLL_3272765080025
// MI455X (gfx1250) — compile-verified
//
#include <hip/hip_runtime.h>
#include <math.h>

#define Q  2048
#define NB 64

typedef float v2f __attribute__((ext_vector_type(2)));
typedef float v8f __attribute__((ext_vector_type(8)));

// ws layout (floats):
//   [0, Q*Q)          S  (becomes L = chol(I + C^1/2 D C^1/2 / sig2e), lower)
//   [Q*Q, +Q)         g  = Z^T m
//   [+Q)              counts (per-location observation counts, as float)
//   [+Q)              b  = C^1/2 D g
//   [+8)              scalars: 0=m^Tm 1=sum r^2 2=g^T D g 3=2*sum log Lii 4=b^T S^-1 b

// ---------------- per-observation statistics --------------------------------
__global__ void obs_stats(const float* __restrict__ yt, const float* __restrict__ yp,
                          const int* __restrict__ zidx,
                          const float* sig2e_p, const float* sig2b_p,
                          float* g, float* counts, float* scal, int n) {
    int i = blockIdx.x * blockDim.x + threadIdx.x;
    if (i >= n) return;
    float sig2 = sig2e_p[0] + sig2b_p[0];
    float r = yt[i] - yp[i];
    float u = 0.5f * (erff(r * rsqrtf(2.0f * sig2)) + 1.0f);
    u = fminf(fmaxf(u, 1e-5f), 1.0f - 1e-5f);
    float m = erfinvf(2.0f * u - 1.0f) * 1.4142135623730951f;
    atomicAdd(&scal[0], m * m);
    atomicAdd(&scal[1], r * r);
    int z = zidx[i];
    atomicAdd(&g[z], m);
    atomicAdd(&counts[z], 1.0f);
}

// ---------------- build S = I + C^1/2 D C^1/2 / sig2e -----------------------
__global__ void build_S(const float* __restrict__ M, const float* __restrict__ counts,
                        const float* sig2e_p, const float* sig2b_p, const float* ell_p,
                        float* __restrict__ S) {
    int idx = blockIdx.x * blockDim.x + threadIdx.x;
    if (idx >= Q * Q) return;
    int i = idx / Q, j = idx - i * Q;
    float sig2e = sig2e_p[0];
    float k = sqrtf(counts[i] * counts[j]) * sig2b_p[0]
            * expf(-M[idx] / (2.0f * ell_p[0]));
    S[idx] = (i == j ? 1.0f : 0.0f) + k / sig2e;
}

// ---------------- w = D g ; b = C^1/2 w ; accumulate g^T D g ----------------
__global__ __launch_bounds__(256) void dg_matvec(const float* __restrict__ M,
                        const float* __restrict__ g, const float* __restrict__ counts,
                        const float* sig2b_p, const float* ell_p,
                        float* __restrict__ b, float* scal) {
    int i = blockIdx.x;
    float inv2l = 1.0f / (2.0f * ell_p[0]);
    float part = 0.0f;
    for (int j = threadIdx.x; j < Q; j += blockDim.x)
        part += expf(-M[i * Q + j] * inv2l) * g[j];
    __shared__ float red[256];
    red[threadIdx.x] = part;
    __syncthreads();
    for (int s = 128; s > 0; s >>= 1) {
        if (threadIdx.x < s) red[threadIdx.x] += red[threadIdx.x + s];
        __syncthreads();
    }
    if (threadIdx.x == 0) {
        float w = sig2b_p[0] * red[0];
        b[i] = sqrtf(counts[i]) * w;
        atomicAdd(&scal[2], g[i] * w);
    }
}

// ---------------- Cholesky: diagonal NB x NB block in LDS -------------------
__global__ __launch_bounds__(NB) void chol_diag(float* __restrict__ S, int k1) {
    __shared__ float A[NB][NB + 1];
    int t = threadIdx.x;
    for (int j = 0; j < NB; ++j) A[t][j] = S[(k1 + t) * Q + k1 + j];
    __syncthreads();
    for (int j = 0; j < NB; ++j) {
        if (t == j) A[j][j] = sqrtf(A[j][j]);
        __syncthreads();
        if (t > j) A[t][j] /= A[j][j];
        __syncthreads();
        if (t > j) {
            float ltj = A[t][j];
            for (int u = j + 1; u <= t; ++u) A[t][u] -= ltj * A[u][j];
        }
        __syncthreads();
    }
    for (int j = 0; j <= t; ++j) S[(k1 + t) * Q + k1 + j] = A[t][j];
}

// ---------------- panel TRSM: L21 = A21 * L11^-T ----------------------------
__global__ __launch_bounds__(128) void trsm_panel(float* __restrict__ S, int k1, int k2) {
    __shared__ float L[NB][NB];
    for (int e = threadIdx.x; e < NB * NB; e += blockDim.x)
        L[e / NB][e % NB] = S[(k1 + e / NB) * Q + k1 + (e % NB)];
    __syncthreads();
    int r = k2 + blockIdx.x * blockDim.x + threadIdx.x;
    if (r >= Q) return;
    float* row = S + r * Q + k1;
    for (int j = 0; j < NB; ++j) {
        float s = row[j];
        for (int u = 0; u < j; ++u) s -= row[u] * L[j][u];
        row[j] = s / L[j][j];
    }
}

// ---------------- trailing SYRK update via V_WMMA_F32_16X16X4_F32 -----------
// One wave per 32x32 macro-tile of the trailing matrix (lower tiles only).
// A fragment (16x4 f32): lanes 0-15 -> M=0..15, K={t,t+1}; lanes 16-31 -> K={t+2,t+3}.
// B fragment (4x16 f32): lanes 0-15 -> N=0..15, K rows {2h, 2h+1} per VGPR half.
// C/D (16x16 f32): VGPR v, lanes 0-15 -> M=v; lanes 16-31 -> M=v+8.
__global__ __launch_bounds__(32) void syrk_wmma(float* __restrict__ S, int k1, int k2) {
    int ti = blockIdx.y, tj = blockIdx.x;
    if (tj > ti) return;                // lower-triangular macro-tiles only (uniform exit)
    int row0 = k2 + ti * 32;
    int col0 = k2 + tj * 32;
    int lane = threadIdx.x;
    int half = lane >> 4;
    int idx  = lane & 15;

    const float* Ar0 = S + (row0 + idx) * Q + k1;
    const float* Ar1 = S + (row0 + 16 + idx) * Q + k1;
    const float* Br0 = S + (col0 + idx) * Q + k1;
    const float* Br1 = S + (col0 + 16 + idx) * Q + k1;

    v8f c00 = {}, c01 = {}, c10 = {}, c11 = {};
#pragma unroll
    for (int t = 0; t < NB; t += 4) {
        int o = t + 2 * half;
        v2f a0 = *(const v2f*)(Ar0 + o);   // 8-byte aligned: k1,t multiples of 4
        v2f a1 = *(const v2f*)(Ar1 + o);
        v2f b0 = *(const v2f*)(Br0 + o);
        v2f b1 = *(const v2f*)(Br1 + o);
        c00 = __builtin_amdgcn_wmma_f32_16x16x4_f32(false, a0, false, b0, (short)0, c00, false, false);
        c01 = __builtin_amdgcn_wmma_f32_16x16x4_f32(false, a0, false, b1, (short)0, c01, false, false);
        c10 = __builtin_amdgcn_wmma_f32_16x16x4_f32(false, a1, false, b0, (short)0, c10, false, false);
        c11 = __builtin_amdgcn_wmma_f32_16x16x4_f32(false, a1, false, b1, (short)0, c11, false, false);
    }
#pragma unroll
    for (int v = 0; v < 8; ++v) {
        int r  = row0 + v + 8 * half;
        int c  = col0 + idx;
        S[r * Q + c]             -= c00[v];
        S[r * Q + c + 16]        -= c01[v];
        S[(r + 16) * Q + c]      -= c10[v];
        S[(r + 16) * Q + c + 16] -= c11[v];
    }
}

// ---------------- 2 * sum log diag(L) ---------------------------------------
__global__ void logdiag(const float* __restrict__ S, float* scal) {
    int i = blockIdx.x * blockDim.x + threadIdx.x;
    if (i >= Q) return;
    atomicAdd(&scal[3], 2.0f * logf(S[i * Q + i]));
}

// ---------------- solve S x = b via L, compute b^T x ------------------------
__global__ __launch_bounds__(1024) void tri_solve(const float* __restrict__ S,
                                                  const float* __restrict__ b,
                                                  float* scal) {
    __shared__ float v[Q];
    __shared__ float red[1024];
    int tid = threadIdx.x, nt = blockDim.x;
    for (int i = tid; i < Q; i += nt) v[i] = 0.0f;
    __syncthreads();
    // forward: L y = b  (left-looking, contiguous row reads)
    for (int i = 0; i < Q; ++i) {
        const float* row = S + i * Q;
        float part = 0.0f;
        for (int j = tid; j < i; j += nt) part += row[j] * v[j];
        red[tid] = part;
        __syncthreads();
        for (int s = nt >> 1; s > 0; s >>= 1) {
            if (tid < s) red[tid] += red[tid + s];
            __syncthreads();
        }
        if (tid == 0) v[i] = (b[i] - red[0]) / row[i];
        __syncthreads();
    }
    // backward: L^T x = y  (right-looking, contiguous row reads)
    for (int i = Q - 1; i >= 0; --i) {
        const float* row = S + i * Q;
        __syncthreads();
        if (tid == 0) v[i] = v[i] / row[i];
        __syncthreads();
        float xi = v[i];
        for (int j = tid; j < i; j += nt) v[j] -= row[j] * xi;
    }
    __syncthreads();
    float part = 0.0f;
    for (int i = tid; i < Q; i += nt) part += b[i] * v[i];
    red[tid] = part;
    __syncthreads();
    for (int s = nt >> 1; s > 0; s >>= 1) {
        if (tid < s) red[tid] += red[tid + s];
        __syncthreads();
    }
    if (tid == 0) scal[4] = red[0];
}

// ---------------- combine scalars -------------------------------------------
__global__ void finalize(const float* scal, const float* sig2e_p, const float* sig2b_p,
                         float* out, int n) {
    float sig2e = sig2e_p[0];
    float sig2  = sig2e + sig2b_p[0];
    float mtm = scal[0], sr2 = scal[1], gDg = scal[2], logS = scal[3], bSb = scal[4];
    // logdet(R) = logdet(V) - N log sig2 ; logdet(V) = N log sig2e + logdet(S)
    float logdetR = (float)n * logf(sig2e / sig2) + logS;
    // m^T V^-1 m via Woodbury, then m^T R^-1 m = sig2 * m^T V^-1 m
    float quad   = gDg - bSb / sig2e;
    float mVinvm = mtm / sig2e - quad / (sig2e * sig2e);
    float mRinvm = sig2 * mVinvm;
    float slp = -0.5f * (float)n * logf(6.283185307179586f * sig2) - sr2 / (2.0f * sig2);
    out[0] = 0.5f * logdetR + 0.5f * mRinvm - 0.5f * mtm + 0.5f * slp;
}

extern "C" void kernel_launch(void* const* d_in, const int* in_sizes, int n_in,
                              void* d_out, int out_size, void* d_ws, size_t ws_size,
                              hipStream_t stream) {
    const float* yt    = (const float*)d_in[0];
    const float* yp    = (const float*)d_in[1];
    const int*   zidx  = (const int*)d_in[2];
    const float* M     = (const float*)d_in[3];
    const float* sig2e = (const float*)d_in[4];
    const float* sig2b = (const float*)d_in[5];
    const float* ell   = (const float*)d_in[6];
    int n = in_sizes[0];

    float* S      = (float*)d_ws;
    float* g      = S + (size_t)Q * Q;
    float* counts = g + Q;
    float* b      = counts + Q;
    float* scal   = b + Q;

    hipMemsetAsync(g, 0, (3 * Q + 8) * sizeof(float), stream);

    obs_stats<<<(n + 255) / 256, 256, 0, stream>>>(yt, yp, zidx, sig2e, sig2b,
                                                   g, counts, scal, n);
    build_S<<<(Q * Q) / 256, 256, 0, stream>>>(M, counts, sig2e, sig2b, ell, S);
    dg_matvec<<<Q, 256, 0, stream>>>(M, g, counts, sig2b, ell, b, scal);

    for (int k = 0; k < Q / NB; ++k) {
        int k1 = k * NB, k2 = k1 + NB;
        chol_diag<<<1, NB, 0, stream>>>(S, k1);
        int rows = Q - k2;
        if (rows > 0) {
            trsm_panel<<<(rows + 127) / 128, 128, 0, stream>>>(S, k1, k2);
            int nt32 = rows / 32;
            syrk_wmma<<<dim3(nt32, nt32), 32, 0, stream>>>(S, k1, k2);
        }
    }

    logdiag<<<Q / 256, 256, 0, stream>>>(S, scal);
    tri_solve<<<1, 1024, 0, stream>>>(S, b, scal);
    finalize<<<1, 1, 0, stream>>>(scal, sig2e, sig2b, (float*)d_out, n);
}